// ReactionTime_77189152244382
// MI455X (gfx1250) — compile-verified
//
#include <hip/hip_runtime.h>
#include <hip/hip_bf16.h>

#define BROWS   4096
#define VCOLS   32000
#define NF4     (VCOLS / 4)                   // 8000 float4 per row
#define CHUNK   1024                          // float4 per staged chunk (16 KB)
#define NCHUNKS ((NF4 + CHUNK - 1) / CHUNK)   // 8 (last chunk = 832)

typedef __attribute__((ext_vector_type(2))) float v2f;
typedef __attribute__((ext_vector_type(8))) float v8f;
typedef __attribute__((ext_vector_type(4))) int v4i;

// ---- fast hardware exp2/log2 ----
#if __has_builtin(__builtin_amdgcn_exp2f)
#define EXP2F(x) __builtin_amdgcn_exp2f(x)
#else
#define EXP2F(x) exp2f(x)
#endif
#if __has_builtin(__builtin_amdgcn_logf)
#define LOG2F(x) __builtin_amdgcn_logf(x)
#else
#define LOG2F(x) log2f(x)
#endif

#if __has_builtin(__builtin_amdgcn_global_load_async_to_lds_b128)
#define HAVE_ASYNC 1
#endif

// ---- async global->LDS copy of 16B, with synchronous fallback ----
__device__ __forceinline__ void cp_async_b128(const float4* g, float4* l) {
#ifdef HAVE_ASYNC
  __builtin_amdgcn_global_load_async_to_lds_b128(
      (__attribute__((address_space(1))) v4i*)((void*)g),
      (__attribute__((address_space(3))) v4i*)((void*)l),
      0, 0);
#else
  *l = *g;
#endif
}

#ifdef HAVE_ASYNC
#if __has_builtin(__builtin_amdgcn_s_wait_asynccnt)
#define WAIT_ASYNC(n) __builtin_amdgcn_s_wait_asynccnt(n)
#else
#define WAIT_ASYNC(n) asm volatile("s_wait_asynccnt " #n ::: "memory")
#endif
#else
#define WAIT_ASYNC(n) do {} while (0)
#endif

// ============================================================================
// Kernel 1: reaction_times pipeline. One block of 1024 threads.
// Bitonic sort 4096 f32 in LDS -> quantiles -> clamp -> interior min/max ->
// normalized rt written to ws.
// ============================================================================
__global__ __launch_bounds__(1024) void rt_kernel(const float* __restrict__ rt,
                                                  float* __restrict__ wrt) {
  __shared__ float sh[BROWS];
  __shared__ float bc[2];
  const int tid = threadIdx.x;

  float4 r4 = ((const float4*)rt)[tid];
  sh[tid * 4 + 0] = r4.x;
  sh[tid * 4 + 1] = r4.y;
  sh[tid * 4 + 2] = r4.z;
  sh[tid * 4 + 3] = r4.w;
  __syncthreads();

  // Bitonic sort ascending
  for (unsigned k = 2; k <= (unsigned)BROWS; k <<= 1) {
    for (unsigned j = k >> 1; j > 0; j >>= 1) {
      for (unsigned i = tid; i < (unsigned)BROWS; i += 1024u) {
        unsigned l = i ^ j;
        if (l > i) {
          float a = sh[i], b = sh[l];
          bool asc = ((i & k) == 0);
          if (asc ? (a > b) : (a < b)) { sh[i] = b; sh[l] = a; }
        }
      }
      __syncthreads();
    }
  }

  if (tid == 0) {
    // linear interpolation: q*(n-1) -> 1023.75 and 3071.25
    bc[0] = sh[1023] + 0.75f * (sh[1024] - sh[1023]);  // lower (q=0.25)
    bc[1] = sh[3071] + 0.25f * (sh[3072] - sh[3071]);  // upper (q=0.75)
  }
  __syncthreads();
  const float lower = bc[0], upper = bc[1];
  __syncthreads();  // done with sorted sh; safe to reuse below

  float r[4] = {r4.x, r4.y, r4.z, r4.w};
  float lmin = 100.0f, lmax = -100.0f;
#pragma unroll
  for (int q = 0; q < 4; ++q) {
    float v = r[q];
    v = (v < lower) ? 0.0f : ((v > upper) ? 1.0f : v);
    r[q] = v;
    if (v != 0.0f && v != 1.0f) {
      lmin = fminf(lmin, v);
      lmax = fmaxf(lmax, v);
    }
  }
  sh[tid] = lmin;
  sh[1024 + tid] = lmax;
  __syncthreads();
  for (int off = 512; off > 0; off >>= 1) {
    if (tid < off) {
      sh[tid] = fminf(sh[tid], sh[tid + off]);
      sh[1024 + tid] = fmaxf(sh[1024 + tid], sh[1024 + tid + off]);
    }
    __syncthreads();
  }
  const float minv = sh[0], maxv = sh[1024];

  float4 o;
  float ov[4];
#pragma unroll
  for (int q = 0; q < 4; ++q) {
    float v = r[q];
    bool interior = (v != 0.0f) && (v != 1.0f);
    ov[q] = interior ? (v - minv) / maxv : v;  // divide by max, per reference
  }
  o.x = ov[0]; o.y = ov[1]; o.z = ov[2]; o.w = ov[3];
  ((float4*)wrt)[tid] = o;
}

// ============================================================================
// Kernel 2: per-row fused argmax + online logsumexp, async-LDS staged.
// One block (256 threads) per row; double-buffered 16KB chunks.
// ============================================================================
__global__ __launch_bounds__(256) void ce_kernel(const float* __restrict__ logits,
                                                 const int* __restrict__ target,
                                                 const float* __restrict__ wrt,
                                                 float* __restrict__ wloss) {
  __shared__ float4 buf[2][CHUNK];
  __shared__ float rm[256], rs[256], rv[256];
  __shared__ int ri[256];

  const int row = blockIdx.x;
  const int tid = threadIdx.x;
  const float4* rowp = (const float4*)(logits + (size_t)row * VCOLS);
  const float L2E = 1.4426950408889634f;

  // stage chunk 0 (uniform 4 issues per lane; tail lanes clamp global src only)
  {
    const int nf = (NF4 < CHUNK) ? NF4 : CHUNK;
#pragma unroll
    for (int k = 0; k < 4; ++k) {
      int f = k * 256 + tid;
      int fg = (f < nf) ? f : (nf - 1);
      cp_async_b128(rowp + fg, &buf[0][f]);
    }
  }

  float m = -3.402823466e38f;
  float s = 0.0f;
  float bestv = -3.402823466e38f;
  int besti = 0;

  for (int c = 0; c < NCHUNKS; ++c) {
    const int base = c * CHUNK;
    if (c + 1 < NCHUNKS) {
      const int base2 = (c + 1) * CHUNK;
      const int nf2 = ((NF4 - base2) < CHUNK) ? (NF4 - base2) : CHUNK;
      float4* dst = &buf[(c + 1) & 1][0];
#pragma unroll
      for (int k = 0; k < 4; ++k) {
        int f = k * 256 + tid;
        int fg = (f < nf2) ? f : (nf2 - 1);
        cp_async_b128(rowp + base2 + fg, dst + f);
      }
      WAIT_ASYNC(4);  // leave next chunk's 4 in flight; chunk c complete
    } else {
      WAIT_ASYNC(0);
    }
    __syncthreads();

    const int nf = ((NF4 - base) < CHUNK) ? (NF4 - base) : CHUNK;
    const float4* bp = &buf[c & 1][0];
#pragma unroll
    for (int k = 0; k < 4; ++k) {
      int f = k * 256 + tid;
      if (f < nf) {
        float4 v4 = bp[f];
        int col = (base + f) * 4;
        float vv[4] = {v4.x, v4.y, v4.z, v4.w};
#pragma unroll
        for (int q = 0; q < 4; ++q) {
          float v = vv[q];
          if (v > bestv) { bestv = v; besti = col + q; }
          if (v <= m) {
            s += EXP2F((v - m) * L2E);
          } else {
            s = s * EXP2F((m - v) * L2E) + 1.0f;  // rare rescale path
            m = v;
          }
        }
      }
    }
    __syncthreads();  // all lanes done reading buf[c&1] before it is re-staged
  }

  // block tree reduction: merge (m,s) pairs and (maxval, first index)
  rm[tid] = m; rs[tid] = s; rv[tid] = bestv; ri[tid] = besti;
  __syncthreads();
  for (int off = 128; off > 0; off >>= 1) {
    if (tid < off) {
      float m1 = rm[tid], s1 = rs[tid];
      float m2 = rm[tid + off], s2 = rs[tid + off];
      float mm = fmaxf(m1, m2);
      rs[tid] = s1 * EXP2F((m1 - mm) * L2E) + s2 * EXP2F((m2 - mm) * L2E);
      rm[tid] = mm;
      float v1 = rv[tid]; int i1 = ri[tid];
      float v2 = rv[tid + off]; int i2 = ri[tid + off];
      if (v2 > v1 || (v2 == v1 && i2 < i1)) { rv[tid] = v2; ri[tid] = i2; }
    }
    __syncthreads();
  }

  if (tid == 0) {
    int tgt = target[row];
    float xt = logits[(size_t)row * VCOLS + tgt];
    float ce = rm[0] - xt + LOG2F(rs[0]) * 0.69314718055994531f;
    wloss[row] = ce + ((ri[0] != tgt) ? wrt[row] : 0.0f);
  }
}

// ============================================================================
// Kernel 3: mean of 4096 per-row losses via V_WMMA_F32_16X16X4_F32 (A = ones:
// every D element is a column-sum of B; sum of D row 0 = total). One wave.
// ============================================================================
__global__ __launch_bounds__(32) void mean_kernel(const float* __restrict__ in,
                                                  float* __restrict__ out) {
  const int lane = threadIdx.x;
  __shared__ float sh[32];
#if __has_builtin(__builtin_amdgcn_wmma_f32_16x16x4_f32)
  v8f c = {0.f, 0.f, 0.f, 0.f, 0.f, 0.f, 0.f, 0.f};
  v2f a;
  a[0] = 1.0f;
  a[1] = 1.0f;
  for (int it = 0; it < BROWS / 64; ++it) {
    const float* p = in + it * 64 + lane * 2;
    v2f b;
    b[0] = p[0];
    b[1] = p[1];
    c = __builtin_amdgcn_wmma_f32_16x16x4_f32(false, a, false, b, (short)0, c,
                                              false, false);
  }
  sh[lane] = c[0];
  __syncthreads();
  if (lane == 0) {
    float t = 0.0f;
    for (int i = 0; i < 16; ++i) t += sh[i];  // D row 0: 16 column sums
    out[0] = t * (1.0f / (float)BROWS);
  }
#else
  float t = 0.0f;
  for (int i = lane; i < BROWS; i += 32) t += in[i];
  sh[lane] = t;
  __syncthreads();
  if (lane == 0) {
    float tt = 0.0f;
    for (int i = 0; i < 32; ++i) tt += sh[i];
    out[0] = tt * (1.0f / (float)BROWS);
  }
#endif
}

// ============================================================================
extern "C" void kernel_launch(void* const* d_in, const int* in_sizes, int n_in,
                              void* d_out, int out_size, void* d_ws,
                              size_t ws_size, hipStream_t stream) {
  (void)in_sizes; (void)n_in; (void)out_size; (void)ws_size;
  const float* logits = (const float*)d_in[0];   // [4096, 32000] f32
  const int*   target = (const int*)d_in[1];     // [4096] int
  const float* rt     = (const float*)d_in[2];   // [4096] f32
  float* w = (float*)d_ws;
  float* wrt   = w;          // 4096 floats: normalized reaction times
  float* wloss = w + BROWS;  // 4096 floats: per-row losses
  float* out = (float*)d_out;

  rt_kernel<<<1, 1024, 0, stream>>>(rt, wrt);
  ce_kernel<<<BROWS, 256, 0, stream>>>(logits, target, wrt, wloss);
  mean_kernel<<<1, 32, 0, stream>>>(wloss, out);
}